// NormalizedCutLossClaude_58557584114372
// MI455X (gfx1250) — compile-verified
//
#include <hip/hip_runtime.h>
#include <hip/hip_bf16.h>

// Normalized-cut loss for MI455X (gfx1250, wave32, WMMA + TDM).
// features: (4,3,64,64) f32, seg_pred: (4,21,64,64) f32 -> scalar f32 loss.
//
// One block = one (batch, 4-row strip): 16 waves, each owning one 4x4 pixel
// tile. The strip's radius-8 neighbor window is a contiguous run of
// runN = rows*64 (<=1280) pixels:
//   - features (3 x runN, channel stride 4096) staged via TENSOR_LOAD_TO_LDS
//     (TDM, D# built per CDNA5 ISA sec.8), waited with s_wait_tensorcnt.
//   - seg staged transposed [m][class] with pad 23 (conflict-free for the 16
//     class-lanes of the WMMA B operand; 2*23=46-dword cluster offset keeps
//     the two half-wave clusters on disjoint banks).
// Hot loop: fused weight w = exp(-ds2/128 - 50*||df||^2) per pair feeds the
// A-matrix of V_WMMA_F32_16X16X4_F32 against B = seg[m,class]; tile0 holds
// classes 0..15, tile1 holds 16..20 plus a ONES column => degree[n] free.
// Tile1 B values are formed branch-free: s1 = fma(seg[m,16+cc], selMul,
// selAdd) so the LDS load is never predicated (no saveexec in the hot loop).

typedef __attribute__((ext_vector_type(2))) float v2f;
typedef __attribute__((ext_vector_type(8))) float v8f;
typedef __attribute__((ext_vector_type(4))) unsigned int v4u;
typedef __attribute__((ext_vector_type(8))) int v8i;
typedef __attribute__((ext_vector_type(4))) int v4i;

#define NB 4
#define NC 3
#define HW 64
#define NPIX 4096
#define NK 21
#define TILES_PER_B 256
#define NTILES 1024
#define SEGP 23          // padded per-pixel class stride in LDS
#define MAXRUN 1280      // 20 rows * 64 cols

// --- TDM: 2D f32 tile (tile_x x tile_y) global->LDS, row stride stride_x ---
__device__ __forceinline__ void tdm_load_2d_f32(void* lds_dst, const float* gsrc,
                                                int tile_x, int tile_y,
                                                int tensor_x, int tensor_y,
                                                int stride_x)
{
  const unsigned long long ga = (unsigned long long)(uintptr_t)gsrc;
  const unsigned int       la = (unsigned int)(uintptr_t)lds_dst; // LDS byte addr
  v4u g0;
  g0[0] = 1u;                                        // count=1, user descriptor
  g0[1] = la;                                        // lds_addr
  g0[2] = (unsigned int)ga;                          // global_addr[31:0]
  g0[3] = (unsigned int)(ga >> 32) | (2u << 30);     // global_addr[56:32]|type=2
  v8i g1;
  g1[0] = (int)(2u << 16);                           // data_size=2 -> 4 bytes
  g1[1] = (int)(((unsigned)tensor_x & 0xFFFFu) << 16);               // tdim0 lo
  g1[2] = (int)(((unsigned)tensor_x >> 16) |
                (((unsigned)tensor_y & 0xFFFFu) << 16));             // tdim0 hi|tdim1 lo
  g1[3] = (int)(((unsigned)tensor_y >> 16) |
                (((unsigned)tile_x & 0xFFFFu) << 16));               // tdim1 hi|tile0
  g1[4] = (int)((unsigned)tile_y & 0xFFFFu);         // tile_dim1 (tile_dim2=0)
  g1[5] = stride_x;                                  // tensor_dim0_stride[31:0]
  g1[6] = 0;                                         // stride[47:32]=0
  g1[7] = 0;
  v4i z4 = {0, 0, 0, 0};                             // 2D tensor: groups 2/3 null
  v8i z8 = {0, 0, 0, 0, 0, 0, 0, 0};                 // clang-23 6-arg form
  __builtin_amdgcn_tensor_load_to_lds(g0, g1, z4, z4, z8, 0);
}

__global__ __launch_bounds__(512) void ncut_tile_kernel(
    const float* __restrict__ F_, const float* __restrict__ S_,
    float* __restrict__ APart, float* __restrict__ VPart)
{
  __shared__ float seg_lds[MAXRUN * SEGP + 32];  // [m][class] padded (+overread)
  __shared__ float feat_lds[NC * MAXRUN];        // [channel][m]

  const int b     = blockIdx.x >> 4;         // 16 strips per batch
  const int strip = blockIdx.x & 15;
  const int bi    = strip << 2;              // strip origin row
  const int wv    = threadIdx.x >> 5;        // wave id = tile column index
  const int lane  = threadIdx.x & 31;
  const int bj    = wv << 2;                 // tile origin col
  const int tile  = blockIdx.x * 16 + wv;    // global tile id (b = tile>>8)

  const float* __restrict__ F = F_ + (size_t)b * NC * NPIX;
  const float* __restrict__ S = S_ + (size_t)b * NK * NPIX;

  // strip neighbor window: image rows [row0, r1) x all 64 cols (contiguous)
  const int row0 = (bi - 8 > 0) ? (bi - 8) : 0;
  const int r1   = (bi + 12 < HW) ? (bi + 12) : HW;
  const int runN = (r1 - row0) << 6;         // <= 1280

  __builtin_prefetch(&S[row0 << 6], 0, 0);   // global_prefetch_b8

  // ---- stage features via Tensor Data Mover (wave 0 issues, EXEC ignored) ----
  if (threadIdx.x < 32) {
    tdm_load_2d_f32(feat_lds, F + (row0 << 6),
                    /*tile   */ runN, NC,
                    /*tensor */ NPIX, NC,
                    /*stride */ NPIX);
    __builtin_amdgcn_s_wait_tensorcnt(0);
  }
  // ---- stage seg transposed [m][class] (coalesced loads, padded stores) ----
  for (int k = 0; k < NK; ++k) {
    const float* __restrict__ src = S + k * NPIX + (row0 << 6);
    for (int x = threadIdx.x; x < runN; x += 512)
      seg_lds[__mul24(x, SEGP) + k] = src[x];
  }
  __syncthreads();

  // ---- per-wave WMMA tile ----
  const int M    = lane & 15;                // A row (n pixel) / B col (class)
  const int half = lane >> 4;                // lanes 16-31 own K=2,3
  const int kkf  = half << 1;
  const int cc   = M;
  // branch-free tile1 B select: s1 = seg[m, 16+cc]*selMul + selAdd
  const float selMul = (cc < 5)  ? 1.0f : 0.0f;   // classes 16..20
  const float selAdd = (cc == 5) ? 1.0f : 0.0f;   // col 21 = ones (degree)

  const int ni = bi + (M >> 2);
  const int nj = bj + (M & 3);
  const int nl = ((ni - row0) << 6) + nj;    // n's LDS index (inside window)
  const float fn0 = feat_lds[nl];
  const float fn1 = feat_lds[runN + nl];
  const float fn2 = feat_lds[2 * runN + nl];

  v8f acc0 = {0.f, 0.f, 0.f, 0.f, 0.f, 0.f, 0.f, 0.f};   // classes 0..15
  v8f acc1 = {0.f, 0.f, 0.f, 0.f, 0.f, 0.f, 0.f, 0.f};   // 16..20 + ones(deg)

  for (int wr = 0; wr < 20; ++wr) {
    const int  mi    = bi - 8 + wr;
    const bool rowok = (mi >= 0) & (mi < HW);
    const int  lr    = (mi - row0) << 6;
    const int  di    = mi - ni;
    const int  di2   = di * di;
#pragma unroll
    for (int wc = 0; wc < 20; wc += 4) {
      const int mjbase = bj - 8 + wc + kkf;
      float wvv[2], s0[2], s1[2];
#pragma unroll
      for (int p = 0; p < 2; ++p) {
        const int  mj   = mjbase + p;
        const bool ok   = rowok & (mj >= 0) & (mj < HW);
        const int  lidx = ok ? (lr + mj) : 0;          // clamp; weight masked
        const float fm0 = feat_lds[lidx];
        const float fm1 = feat_lds[runN + lidx];
        const float fm2 = feat_lds[2 * runN + lidx];
        const float d0 = fn0 - fm0;
        const float d1 = fn1 - fm1;
        const float d2 = fn2 - fm2;
        const float fd = d0 * d0 + d1 * d1 + d2 * d2;
        const int   dj  = mj - nj;
        const int   ds2 = di2 + dj * dj;
        const float arg = (float)ds2 * (-1.0f / 128.0f) - 50.0f * fd;
        const float e   = __expf(arg);                 // one v_exp per pair
        wvv[p] = (ok && (ds2 <= 64)) ? e : 0.0f;       // RADIUS^2 cutoff
        // two seg loads off one base (dword offsets 0 / 16) -> ds_load_2addr
        const float* rowp = &seg_lds[__mul24(lidx, SEGP) + cc];
        s0[p] = rowp[0];                               // class cc
        const float sv = rowp[16];                     // class 16+cc (overread
        s1[p] = fmaf(sv, selMul, selAdd);              //  x0 for cc>4)
      }
      v2f A  = { wvv[0], wvv[1] };
      v2f B0 = { s0[0], s0[1] };
      v2f B1 = { s1[0], s1[1] };
      acc0 = __builtin_amdgcn_wmma_f32_16x16x4_f32(
                 false, A, false, B0, (short)0, acc0, false, false);
      acc1 = __builtin_amdgcn_wmma_f32_16x16x4_f32(
                 false, A, false, B1, (short)0, acc1, false, false);
    }
  }

  // ---- epilogue: assoc_k = sum_n seg[n,k]*D[n,k]; vol_k = sum_n seg[n,k]*deg[n]
  // D layout: lane holds col cc, rows M = half*8+r. degree = D1 col 5
  // -> rows 0..7 in lane 5, rows 8..15 in lane 21.
  float sa0 = 0.f, sv0 = 0.f, sa1 = 0.f, sv1 = 0.f;
#pragma unroll
  for (int r = 0; r < 8; ++r) {
    const int   Mr   = (half << 3) + r;
    const int   nri  = bi + (Mr >> 2);
    const int   nrj  = bj + (Mr & 3);
    const int   nlid = ((nri - row0) << 6) + nrj;
    const float deg  = __shfl(acc1[r], half ? 21 : 5, 32);
    const float* rowp = &seg_lds[__mul24(nlid, SEGP) + cc];
    const float sc0  = rowp[0];
    sa0 += sc0 * acc0[r];
    sv0 += sc0 * deg;
    const float sc1  = rowp[16] * selMul;              // classes 16..20 only
    sa1 += sc1 * acc1[r];
    sv1 += sc1 * deg;
  }
  sa0 += __shfl_xor(sa0, 16, 32);
  sv0 += __shfl_xor(sv0, 16, 32);
  sa1 += __shfl_xor(sa1, 16, 32);
  sv1 += __shfl_xor(sv1, 16, 32);

  if (lane < 16) {                           // one writer per (tile, class)
    APart[tile * NK + cc] = sa0;
    VPart[tile * NK + cc] = sv0;
    if (cc < 5) {
      APart[tile * NK + 16 + cc] = sa1;
      VPart[tile * NK + 16 + cc] = sv1;
    }
  }
}

// Deterministic reduction of per-tile partials -> scalar loss.
__global__ void ncut_finalize_kernel(const float* __restrict__ APart,
                                     const float* __restrict__ VPart,
                                     float* __restrict__ out)
{
  __shared__ float red[NB * NK];
  const int t = threadIdx.x;
  if (t < NB * NK) {
    const int b = t / NK, k = t % NK;
    float a = 0.f, v = 0.f;
    for (int i = 0; i < TILES_PER_B; ++i) {
      const int tile = b * TILES_PER_B + i;
      a += APart[tile * NK + k];
      v += VPart[tile * NK + k];
    }
    red[t] = a / (v + 1e-8f);
  }
  __syncthreads();
  if (t == 0) {
    float s = 0.f;
    for (int i = 0; i < NB * NK; ++i) s += red[i];
    out[0] = -s * (1.0f / (float)(NB * NK));
  }
}

extern "C" void kernel_launch(void* const* d_in, const int* in_sizes, int n_in,
                              void* d_out, int out_size, void* d_ws, size_t ws_size,
                              hipStream_t stream) {
  const float* features = (const float*)d_in[0];   // (4,3,64,64) f32
  const float* seg_pred = (const float*)d_in[1];   // (4,21,64,64) f32
  float* APart = (float*)d_ws;                     // NTILES*NK floats
  float* VPart = APart + NTILES * NK;              // NTILES*NK floats
  float* out   = (float*)d_out;                    // scalar f32

  ncut_tile_kernel<<<dim3(NB * 16), dim3(512), 0, stream>>>(
      features, seg_pred, APart, VPart);
  ncut_finalize_kernel<<<dim3(1), dim3(128), 0, stream>>>(APart, VPart, out);
}